// WeightedGCNEncoder_27118423507684
// MI455X (gfx1250) — compile-verified
//
#include <hip/hip_runtime.h>
#include <hip/hip_bf16.h>

typedef __attribute__((ext_vector_type(2))) float v2f;
typedef __attribute__((ext_vector_type(8))) float v8f;

#define D 64

// ---------------------------------------------------------------------------
// deg[row[e]] += w[e]   (native f32 atomic on gfx1250)
// ---------------------------------------------------------------------------
__global__ void gcn_degree_kernel(const long long* __restrict__ row,
                                  const float* __restrict__ w,
                                  float* __restrict__ deg, int E) {
    int e = blockIdx.x * blockDim.x + threadIdx.x;
    if (e < E) {
        unsafeAtomicAdd(&deg[(int)row[e]], w[e]);
    }
}

// ---------------------------------------------------------------------------
// dinv[n] = deg[n] > 0 ? rsqrt(deg[n]) : 0
// ---------------------------------------------------------------------------
__global__ void gcn_dinv_kernel(const float* __restrict__ deg,
                                float* __restrict__ dinv, int N) {
    int n = blockIdx.x * blockDim.x + threadIdx.x;
    if (n < N) {
        float d = deg[n];
        dinv[n] = (d > 0.0f) ? rsqrtf(d) : 0.0f;
    }
}

// ---------------------------------------------------------------------------
// H = (applyRelu ? relu(A) : A) @ W + b     A:[N,64] W:[64,64] b:[64]
// Block = 128 threads = 4 waves. Block owns a 16-row strip; wave i owns the
// 16-column tile [16*i, 16*i+16). K=64 swept as 16 chained
// V_WMMA_F32_16X16X4_F32 ops. W and the A strip are staged in LDS.
// ---------------------------------------------------------------------------
__global__ void __launch_bounds__(128)
gcn_gemm_wmma_kernel(const float* __restrict__ A,
                     const float* __restrict__ W,
                     const float* __restrict__ bias,
                     float* __restrict__ H,
                     int N, int applyRelu) {
    __shared__ float sW[D * D];     // 16 KB
    __shared__ float sA[16 * D];    // 4 KB

    const int tid = threadIdx.x;
    const int m0  = blockIdx.x * 16;

    // Stage W: 4096 elements over 128 threads.
    for (int i = tid; i < D * D; i += 128) sW[i] = W[i];
    // Stage 16x64 A strip (apply ReLU here for layer 2).
    for (int i = tid; i < 16 * D; i += 128) {
        int r = i >> 6, c = i & 63;
        int gr = m0 + r;
        float v = (gr < N) ? A[gr * D + c] : 0.0f;
        if (applyRelu) v = fmaxf(v, 0.0f);
        sA[i] = v;
    }
    __syncthreads();

    const int wave  = tid >> 5;        // 0..3 -> column tile
    const int lane  = tid & 31;
    const int n0    = wave * 16;
    const int lmod  = lane & 15;       // M index (A) / N index (B,C)
    const int khalf = (lane >> 4) * 2; // lanes 16..31 hold K+2, K+3

    v8f acc = {};
    #pragma unroll
    for (int k0 = 0; k0 < D; k0 += 4) {
        // A-frag: VGPR0 = {K=k0 | K=k0+2}, VGPR1 = {K=k0+1 | K=k0+3}
        v2f a;
        a.x = sA[lmod * D + k0 + khalf + 0];
        a.y = sA[lmod * D + k0 + khalf + 1];
        // B-frag: same K split, row-striped across lanes.
        v2f b;
        b.x = sW[(k0 + khalf + 0) * D + n0 + lmod];
        b.y = sW[(k0 + khalf + 1) * D + n0 + lmod];
        acc = __builtin_amdgcn_wmma_f32_16x16x4_f32(
            /*neg_a=*/false, a, /*neg_b=*/false, b,
            /*c_mod=*/(short)0, acc, /*reuse_a=*/false, /*reuse_b=*/false);
    }

    // D-matrix: VGPR v -> M = v + (lane<16 ? 0 : 8), N = n0 + lmod.
    const int mbase = m0 + ((lane >> 4) << 3);
    const float bn = bias[n0 + lmod];
    #pragma unroll
    for (int v = 0; v < 8; ++v) {
        int gm = mbase + v;
        if (gm < N) H[gm * D + n0 + lmod] = acc[v] + bn;
    }
}

// ---------------------------------------------------------------------------
// out[col[e], :] += dinv[row[e]] * w[e] * dinv[col[e]] * H[row[e], :]
// One wave32 per edge; each lane owns a float2 slice of the 64-wide row.
// ---------------------------------------------------------------------------
__global__ void gcn_spmm_kernel(const long long* __restrict__ row,
                                const long long* __restrict__ col,
                                const float* __restrict__ w,
                                const float* __restrict__ dinv,
                                const float* __restrict__ H,
                                float* __restrict__ out, int E) {
    int gid  = blockIdx.x * blockDim.x + threadIdx.x;
    int e    = gid >> 5;
    int lane = gid & 31;
    if (e >= E) return;

    int r = (int)row[e];
    int c = (int)col[e];
    float norm = dinv[r] * w[e] * dinv[c];

    const float2* hp = (const float2*)(H + (size_t)r * D);
    float2 v = hp[lane];
    float* op = out + (size_t)c * D + 2 * lane;
    unsafeAtomicAdd(op + 0, norm * v.x);
    unsafeAtomicAdd(op + 1, norm * v.y);
}

// ---------------------------------------------------------------------------
// Inputs: 0=x[N,64] f32, 1=edge_index[2,E] i64, 2=edge_weight[E] f32,
//         3=W1[64,64], 4=b1[64], 5=W2[64,64], 6=b2[64]
// Output: [N,64] f32
// ---------------------------------------------------------------------------
extern "C" void kernel_launch(void* const* d_in, const int* in_sizes, int n_in,
                              void* d_out, int out_size, void* d_ws, size_t ws_size,
                              hipStream_t stream) {
    const float*     x    = (const float*)d_in[0];
    const long long* eidx = (const long long*)d_in[1];
    const float*     ew   = (const float*)d_in[2];
    const float*     W1   = (const float*)d_in[3];
    const float*     b1   = (const float*)d_in[4];
    const float*     W2   = (const float*)d_in[5];
    const float*     b2   = (const float*)d_in[6];

    const int N = in_sizes[0] / D;
    const int E = in_sizes[1] / 2;
    const long long* erow = eidx;
    const long long* ecol = eidx + E;

    float* deg  = (float*)d_ws;                     // [N]
    float* dinv = deg  + N;                         // [N]
    float* h    = dinv + N;                         // [N,64] (h1, then h2)
    float* agg  = h    + (size_t)N * D;             // [N,64] (relu input of L2)
    float* out  = (float*)d_out;                    // [N,64]

    // Zero the accumulation targets (capture-safe async memsets).
    hipMemsetAsync(deg, 0, (size_t)N * sizeof(float), stream);
    hipMemsetAsync(agg, 0, (size_t)N * D * sizeof(float), stream);
    hipMemsetAsync(out, 0, (size_t)N * D * sizeof(float), stream);

    // Normalization coefficients.
    gcn_degree_kernel<<<(E + 255) / 256, 256, 0, stream>>>(erow, ew, deg, E);
    gcn_dinv_kernel<<<(N + 255) / 256, 256, 0, stream>>>(deg, dinv, N);

    const int gemmBlocks = (N + 15) / 16;
    const long long spmmThreads = (long long)E * 32;
    const int spmmBlocks = (int)((spmmThreads + 255) / 256);

    // Layer 1: h1 = x@W1 + b1 ; agg = scatter(norm * h1)
    gcn_gemm_wmma_kernel<<<gemmBlocks, 128, 0, stream>>>(x, W1, b1, h, N, 0);
    gcn_spmm_kernel<<<spmmBlocks, 256, 0, stream>>>(erow, ecol, ew, dinv, h, agg, E);

    // Layer 2: h2 = relu(agg)@W2 + b2 ; out = scatter(norm * h2)
    gcn_gemm_wmma_kernel<<<gemmBlocks, 128, 0, stream>>>(agg, W2, b2, h, N, 1);
    gcn_spmm_kernel<<<spmmBlocks, 256, 0, stream>>>(erow, ecol, ew, dinv, h, out, E);
}